// ssim_loss_90013924589948
// MI455X (gfx1250) — compile-verified
//
#include <hip/hip_runtime.h>

// SSIM loss for (16,512,512) fp32 pairs, replicating the reference's 1x11
// Gaussian conv with (5,5) padding (output 16x3x522x512; the 10 zero-pad rows
// give ssim==1, and the 3 identical channels cancel in the mean).
//
// The 11-tap horizontal conv is a banded-Toeplitz matmul executed with
// V_WMMA_F32_16X16X4_F32: per 16x16 output tile, 7 K-steps of 4 over a
// 16x28 input slab, for 5 convolved quantities (x1, x2, x1^2, x2^2, x1*x2).
// Each wave processes 4 adjacent tiles to amortize setup + reduction.

typedef float v2f __attribute__((ext_vector_type(2)));
typedef float v8f __attribute__((ext_vector_type(8)));

#define IMG_W   512
#define IMG_H   512
#define IMG_B   16
#define TILES_PER_WAVE 4
#define WAVES_PER_BLK  8
#define TOTAL_WAVES    (IMG_B * 32 * 8)            // 16 batch * 32 rowtiles * 8 colgroups = 4096
#define NUM_BLOCKS     (TOTAL_WAVES / WAVES_PER_BLK) // 512

// Normalized Gaussian weights (sigma=1.5, 11 taps), bit-exact to the
// expf->f32 / f32-sum normalization the compiler folded in round 1:
//   0x3a86cab8 0x3bf8ff02 0x3d13758c 0x3ddff880 0x3e5a1e20 0x3e8832b1 (center)
constexpr float GW[11] = {
    0x86CAB8p-33f, 0xF8FF02p-31f, 0x93758Cp-28f, 0xDFF880p-27f, 0xDA1E20p-26f,
    0x8832B1p-25f,
    0xDA1E20p-26f, 0xDFF880p-27f, 0x93758Cp-28f, 0xF8FF02p-31f, 0x86CAB8p-33f
};

// Per-lane WMMA B fragments for the 7 K-steps, built at compile time.
// B[k][n] = GW[k-n] for 0<=k-n<=10 else 0.
// Lane layout (mirror of the documented A layout): lanes 0-15 hold K=4s+0,4s+1
// for column n=lane; lanes 16-31 hold K=4s+2,4s+3 for column n=lane-16.
struct alignas(16) BFTab { float v[7][32][2]; };

constexpr BFTab make_bftab() {
    BFTab t{};
    for (int s = 0; s < 7; ++s)
        for (int l = 0; l < 32; ++l) {
            const int n  = l & 15;
            const int k0 = 4 * s + ((l >> 4) << 1);
            for (int j = 0; j < 2; ++j) {
                const int d = k0 + j - n;
                t.v[s][l][j] = (d >= 0 && d <= 10) ? GW[d] : 0.0f;
            }
        }
    return t;
}
__device__ __constant__ BFTab BF_TAB = make_bftab();

__global__ __launch_bounds__(256) void ssim_tile_kernel(
    const float* __restrict__ img1,
    const float* __restrict__ img2,
    float* __restrict__ partial)
{
    const int lane = threadIdx.x & 31;
    // Force wave-uniform values into SGPRs so tile selection and the
    // edge-vs-interior branch are scalar (EXEC stays all-1s around WMMA).
    const int warp = __builtin_amdgcn_readfirstlane(threadIdx.x >> 5);
    const int wg   = blockIdx.x * WAVES_PER_BLK + warp;   // 0..4095

    const int b      = wg >> 8;          // batch (256 waves per image)
    const int rem    = wg & 255;
    const int r0     = (rem >> 3) << 4;  // tile row start
    const int c0base = (rem & 7) << 6;   // first of 4 tiles: cq*64

    const int n      = lane & 15;        // A row / B column selector
    const int kb_off = (lane >> 4) << 1; // K sub-offset: 0 or 2

    // Load the 7 precomputed B fragments (global_load_b64 from __constant__).
    v2f bf[7];
#pragma unroll
    for (int s = 0; s < 7; ++s)
        bf[s] = *reinterpret_cast<const v2f*>(&BF_TAB.v[s][lane][0]);

    const int row = r0 + n;
    const size_t rbase = ((size_t)(b * IMG_H + row)) << 9;  // *512
    const float* __restrict__ p1 = img1 + rbase;
    const float* __restrict__ p2 = img2 + rbase;

    const float C1 = 0.0001f;   // (0.01*1)^2
    const float C2 = 0.0009f;   // (0.03*1)^2
    float part = 0.f;

#pragma unroll 1
    for (int tq = 0; tq < TILES_PER_WAVE; ++tq) {
        const int c0 = c0base + (tq << 4);

        v8f acc_mu1 = {0.f,0.f,0.f,0.f,0.f,0.f,0.f,0.f};
        v8f acc_mu2 = {0.f,0.f,0.f,0.f,0.f,0.f,0.f,0.f};
        v8f acc_m11 = {0.f,0.f,0.f,0.f,0.f,0.f,0.f,0.f};
        v8f acc_m22 = {0.f,0.f,0.f,0.f,0.f,0.f,0.f,0.f};
        v8f acc_m12 = {0.f,0.f,0.f,0.f,0.f,0.f,0.f,0.f};

        if (c0 != 0 && c0 != (IMG_W - 16)) {
            // Interior tile: all 28 K columns in range, no masking.
            const float* __restrict__ q1 = p1 + (c0 - 5) + kb_off;
            const float* __restrict__ q2 = p2 + (c0 - 5) + kb_off;
#pragma unroll
            for (int s = 0; s < 7; ++s) {
                const int o = s << 2;
                const float x1a = q1[o], x1b = q1[o + 1];
                const float x2a = q2[o], x2b = q2[o + 1];
                v2f a1, a2, a11, a22, a12;
                a1[0]  = x1a;       a1[1]  = x1b;
                a2[0]  = x2a;       a2[1]  = x2b;
                a11[0] = x1a * x1a; a11[1] = x1b * x1b;
                a22[0] = x2a * x2a; a22[1] = x2b * x2b;
                a12[0] = x1a * x2a; a12[1] = x1b * x2b;
                const v2f bb = bf[s];
                acc_mu1 = __builtin_amdgcn_wmma_f32_16x16x4_f32(false, a1,  false, bb, (short)0, acc_mu1, false, false);
                acc_mu2 = __builtin_amdgcn_wmma_f32_16x16x4_f32(false, a2,  false, bb, (short)0, acc_mu2, false, false);
                acc_m11 = __builtin_amdgcn_wmma_f32_16x16x4_f32(false, a11, false, bb, (short)0, acc_m11, false, false);
                acc_m22 = __builtin_amdgcn_wmma_f32_16x16x4_f32(false, a22, false, bb, (short)0, acc_m22, false, false);
                acc_m12 = __builtin_amdgcn_wmma_f32_16x16x4_f32(false, a12, false, bb, (short)0, acc_m12, false, false);
            }
        } else {
            // Edge tile: clamp addresses, zero out-of-range contributions.
#pragma unroll
            for (int s = 0; s < 7; ++s) {
                const int col = c0 - 5 + (s << 2) + kb_off;
                const int cA  = col     < 0 ? 0 : (col     > 511 ? 511 : col);
                const int cB  = col + 1 < 0 ? 0 : (col + 1 > 511 ? 511 : col + 1);
                const bool vA = (unsigned)col       < (unsigned)IMG_W;
                const bool vB = (unsigned)(col + 1) < (unsigned)IMG_W;
                const float x1a = vA ? p1[cA] : 0.f;
                const float x1b = vB ? p1[cB] : 0.f;
                const float x2a = vA ? p2[cA] : 0.f;
                const float x2b = vB ? p2[cB] : 0.f;
                v2f a1, a2, a11, a22, a12;
                a1[0]  = x1a;       a1[1]  = x1b;
                a2[0]  = x2a;       a2[1]  = x2b;
                a11[0] = x1a * x1a; a11[1] = x1b * x1b;
                a22[0] = x2a * x2a; a22[1] = x2b * x2b;
                a12[0] = x1a * x2a; a12[1] = x1b * x2b;
                const v2f bb = bf[s];
                acc_mu1 = __builtin_amdgcn_wmma_f32_16x16x4_f32(false, a1,  false, bb, (short)0, acc_mu1, false, false);
                acc_mu2 = __builtin_amdgcn_wmma_f32_16x16x4_f32(false, a2,  false, bb, (short)0, acc_mu2, false, false);
                acc_m11 = __builtin_amdgcn_wmma_f32_16x16x4_f32(false, a11, false, bb, (short)0, acc_m11, false, false);
                acc_m22 = __builtin_amdgcn_wmma_f32_16x16x4_f32(false, a22, false, bb, (short)0, acc_m22, false, false);
                acc_m12 = __builtin_amdgcn_wmma_f32_16x16x4_f32(false, a12, false, bb, (short)0, acc_m12, false, false);
            }
        }

        // Pointwise SSIM on the 8 accumulator elements this lane owns.
        // (The tile sum is layout-invariant, so the C/D mapping is irrelevant.)
#pragma unroll
        for (int e = 0; e < 8; ++e) {
            const float mu1 = acc_mu1[e];
            const float mu2 = acc_mu2[e];
            const float m1s = mu1 * mu1;
            const float m2s = mu2 * mu2;
            const float m12 = mu1 * mu2;
            const float s1  = acc_m11[e] - m1s;
            const float s2  = acc_m22[e] - m2s;
            const float s12 = acc_m12[e] - m12;
            const float num = __builtin_fmaf(2.f, m12, C1) * __builtin_fmaf(2.f, s12, C2);
            const float den = (m1s + m2s + C1) * (s1 + s2 + C2);
            part += num * __builtin_amdgcn_rcpf(den);
        }
    }

    // wave32 reduction (deterministic)
#pragma unroll
    for (int off = 16; off > 0; off >>= 1)
        part += __shfl_down(part, off, 32);

    __shared__ float sm[WAVES_PER_BLK];
    if (lane == 0) sm[warp] = part;
    __syncthreads();
    if (threadIdx.x == 0) {
        float s = 0.f;
#pragma unroll
        for (int i = 0; i < WAVES_PER_BLK; ++i) s += sm[i];
        partial[blockIdx.x] = s;
    }
}

__global__ __launch_bounds__(256) void ssim_finalize_kernel(
    const float* __restrict__ partial, float* __restrict__ out)
{
    __shared__ float sm[256];
    float s = 0.f;
    for (int i = threadIdx.x; i < NUM_BLOCKS; i += 256) s += partial[i];
    sm[threadIdx.x] = s;
    __syncthreads();
    for (int off = 128; off > 0; off >>= 1) {
        if (threadIdx.x < off) sm[threadIdx.x] += sm[threadIdx.x + off];
        __syncthreads();
    }
    if (threadIdx.x == 0) {
        // H-pad rows (KH=1, pad 5): 10 rows x 512 cols x 16 batch, ssim==1
        const float pad_sum = 81920.f;           // 16*10*512
        const float denom   = 4276224.f;         // 16*522*512
        out[0] = 1.f - (sm[0] + pad_sum) / denom;
    }
}

extern "C" void kernel_launch(void* const* d_in, const int* in_sizes, int n_in,
                              void* d_out, int out_size, void* d_ws, size_t ws_size,
                              hipStream_t stream) {
    (void)in_sizes; (void)n_in; (void)out_size; (void)ws_size;
    const float* img1 = (const float*)d_in[0];
    const float* img2 = (const float*)d_in[1];
    float* ws  = (float*)d_ws;           // needs NUM_BLOCKS*4 = 2 KB
    float* out = (float*)d_out;

    ssim_tile_kernel<<<NUM_BLOCKS, 256, 0, stream>>>(img1, img2, ws);
    ssim_finalize_kernel<<<1, 256, 0, stream>>>(ws, out);
}